// ModulatedFromRGB_31138512896103
// MI455X (gfx1250) — compile-verified
//
#include <hip/hip_runtime.h>
#include <hip/hip_bf16.h>

typedef float v2f __attribute__((ext_vector_type(2)));
typedef float v4f __attribute__((ext_vector_type(4)));
typedef float v8f __attribute__((ext_vector_type(8)));

// ---------------------------------------------------------------------------
// Kernel A: fused InverseHaar(up2) -> blur(down2) -> Haar(down2)
// x [16,12,256,256] -> inp [16,12,128,128]
// Composite separable 1D kernels (input filter a in {l,h}, output filter c):
//   (l,l):[1,7,7,1]/16 (l,h):[1,5,-5,-1]/16 (h,l):[1,1,-1,-1]/16 (h,h):[1,-1,-1,1]/16
// Taps cover x[2m-1 .. 2m+2] per dim, zero-extended at borders (exact).
// ---------------------------------------------------------------------------
__global__ void __launch_bounds__(256) fused_fir_kernel(
    const float* __restrict__ x, float* __restrict__ inp) {
  const float cW[2][2][4] = {
      {{1.f/16, 7.f/16, 7.f/16, 1.f/16},   // a=l, c=l
       {1.f/16, 5.f/16, -5.f/16, -1.f/16}},// a=l, c=h
      {{1.f/16, 1.f/16, -1.f/16, -1.f/16}, // a=h, c=l
       {1.f/16, -1.f/16, -1.f/16, 1.f/16}} // a=h, c=h
  };

  int t = blockIdx.x * blockDim.x + threadIdx.x;   // 786432 threads total
  int mx = t & 127;
  int my = (t >> 7) & 127;
  int rest = t >> 14;          // b*3 + color, 48 values
  int color = rest % 3;
  int b = rest / 3;

  int y0 = 2 * my - 1;
  int x0 = 2 * mx - 1;

  float acc0 = 0.f, acc1 = 0.f, acc2 = 0.f, acc3 = 0.f;  // ll, lh, hl, hh out

  #pragma unroll
  for (int ib = 0; ib < 4; ++ib) {
    // band -> (vertical filter av, horizontal filter ah); ll=0,lh=1,hl=2,hh=3
    const int av = ib & 1;
    const int ah = (ib >> 1) & 1;
    const float sgn = (ib == 1 || ib == 2) ? -1.f : 1.f;
    const float* __restrict__ xb =
        x + (((size_t)(b * 12 + ib * 3 + color)) << 16);  // 256*256 plane

    float rowL[4], rowH[4];
    #pragma unroll
    for (int dy = 0; dy < 4; ++dy) {
      int yy = y0 + dy;
      bool vy = ((unsigned)yy < 256u);
      const float* __restrict__ row = xb + (size_t)yy * 256;
      float rl = 0.f, rh = 0.f;
      #pragma unroll
      for (int dx = 0; dx < 4; ++dx) {
        int xx = x0 + dx;
        float v = (vy && ((unsigned)xx < 256u)) ? row[xx] : 0.f;
        rl += cW[ah][0][dx] * v;
        rh += cW[ah][1][dx] * v;
      }
      rowL[dy] = rl;
      rowH[dy] = rh;
    }
    float cLL = 0.f, cHL = 0.f, cLH = 0.f, cHH = 0.f;
    #pragma unroll
    for (int dy = 0; dy < 4; ++dy) {
      cLL += cW[av][0][dy] * rowL[dy];   // out (v=l, h=l) = ll
      cHL += cW[av][1][dy] * rowL[dy];   // out (v=h, h=l) = lh
      cLH += cW[av][0][dy] * rowH[dy];   // out (v=l, h=h) = hl
      cHH += cW[av][1][dy] * rowH[dy];   // out (v=h, h=h) = hh
    }
    acc0 += sgn * cLL;
    acc1 += sgn * cHL;
    acc2 += sgn * cLH;
    acc3 += sgn * cHH;
  }

  size_t obase = ((size_t)(b * 12) << 14) + ((size_t)my << 7) + mx; // 128*128 planes
  inp[obase + ((size_t)(0 * 3 + color) << 14)] = acc0;
  inp[obase + ((size_t)(1 * 3 + color) << 14)] = acc1;
  inp[obase + ((size_t)(2 * 3 + color) << 14)] = acc2;
  inp[obase + ((size_t)(3 * 3 + color) << 14)] = acc3;
}

// ---------------------------------------------------------------------------
// Kernel B: 1x1 conv (12 -> 256) as WMMA f32 16x16x4 GEMM + bias + lrelu*sqrt2
// inp [16,12,128,128] , W [256,12] scaled by 1/sqrt(12) -> out [16,256,128,128]
// A frag (16x4 f32): lane<16 M=lane {K0,K1}, lane>=16 M=lane-16 {K2,K3}
// B frag (4x16 f32): lane<16 N=lane {K0,K1}, lane>=16 N=lane-16 {K2,K3}
// D (16x16 f32): vgpr r: lanes0-15 M=r, lanes16-31 M=r+8, N=lane&15
// ---------------------------------------------------------------------------
__global__ void __launch_bounds__(128, 1) gemm1x1_lrelu_kernel(
    const float* __restrict__ inp, const float* __restrict__ Wt,
    const float* __restrict__ bias, float* __restrict__ out, int totalWaves) {
  const int lane = threadIdx.x & 31;
  const int lane16 = lane & 15;
  const bool hi = lane >= 16;
  const float scale = 0.288675134594812882f;  // 1/sqrt(12)
  const float sqrt2 = 1.41421356237309515f;

  // Preload all A fragments (weights): 16 M-tiles x 3 K-steps, 2 VGPRs each
  v2f A[16][3];
  #pragma unroll
  for (int mt = 0; mt < 16; ++mt) {
    #pragma unroll
    for (int ks = 0; ks < 3; ++ks) {
      int o = mt * 16 + lane16;
      int c = ks * 4 + (hi ? 2 : 0);
      A[mt][ks].x = Wt[o * 12 + c] * scale;
      A[mt][ks].y = Wt[o * 12 + c + 1] * scale;
    }
  }

  const int waveId = blockIdx.x * (blockDim.x >> 5) + (threadIdx.x >> 5);
  const int strips = 16 * 1024;  // 16 batches * (16384 pixels / 16)

  for (int s = waveId; s < strips; s += totalWaves) {
    int b = s >> 10;
    int p0 = (s & 1023) << 4;

    const float* __restrict__ ibase =
        inp + ((size_t)(b * 12) << 14) + p0 + lane16;
    v2f Bf[3];
    #pragma unroll
    for (int ks = 0; ks < 3; ++ks) {
      int c = ks * 4 + (hi ? 2 : 0);
      Bf[ks].x = ibase[(size_t)c << 14];
      Bf[ks].y = ibase[(size_t)(c + 1) << 14];
    }

    float* __restrict__ obase = out + ((size_t)(b * 256) << 14) + p0 + lane16;

    #pragma unroll
    for (int mt = 0; mt < 16; ++mt) {
      v8f acc = {};
      acc = __builtin_amdgcn_wmma_f32_16x16x4_f32(
          false, A[mt][0], false, Bf[0], (short)0, acc, false, false);
      acc = __builtin_amdgcn_wmma_f32_16x16x4_f32(
          false, A[mt][1], false, Bf[1], (short)0, acc, false, false);
      acc = __builtin_amdgcn_wmma_f32_16x16x4_f32(
          false, A[mt][2], false, Bf[2], (short)0, acc, false, false);

      const int mbase = mt * 16 + (hi ? 8 : 0);
      v4f b0 = *(const v4f*)(bias + mbase);
      v4f b1 = *(const v4f*)(bias + mbase + 4);
      #pragma unroll
      for (int r = 0; r < 8; ++r) {
        float v = acc[r] + (r < 4 ? b0[r] : b1[r - 4]);
        v = (v > 0.f ? v : 0.2f * v) * sqrt2;
        obase[((size_t)(mbase + r)) << 14] = v;
      }
    }
  }
}

extern "C" void kernel_launch(void* const* d_in, const int* in_sizes, int n_in,
                              void* d_out, int out_size, void* d_ws, size_t ws_size,
                              hipStream_t stream) {
  const float* x  = (const float*)d_in[0];   // [16,12,256,256]
  const float* W  = (const float*)d_in[1];   // [256,12,1,1]
  const float* bb = (const float*)d_in[2];   // [256]

  float* inp = (float*)d_out;                          // [16,12,128,128]
  float* out = (float*)d_out + (size_t)16 * 12 * 128 * 128;  // [16,256,128,128]

  // Kernel A: 16*3*128*128 = 786432 threads
  fused_fir_kernel<<<3072, 256, 0, stream>>>(x, inp);

  // Kernel B: 512 blocks * 4 waves = 2048 waves, 16384 strips -> 8 each (uniform)
  const int blocks = 512;
  const int totalWaves = blocks * 4;
  gemm1x1_lrelu_kernel<<<blocks, 128, 0, stream>>>(inp, W, bb, out, totalWaves);
}